// SlotAttention_6030134083602
// MI455X (gfx1250) — compile-verified
//
#include <hip/hip_runtime.h>
#include <hip/hip_bf16.h>

#define B_ 128
#define P_ 4096
#define F_ 32
#define K_ 8
#define D_ 64
#define SCALE_ 0.125f
#define EPS_ATTN_ 1e-8f
#define EPS_LN_ 1e-5f

typedef __attribute__((ext_vector_type(2))) float v2f;
typedef __attribute__((ext_vector_type(8))) float v8f;

// D = A(16x4) * B(4x16) + C, full f32 WMMA (gfx1250)
#define WMMA_F32_16X16X4(A, Bf, C) \
  __builtin_amdgcn_wmma_f32_16x16x4_f32(false, (A), false, (Bf), (short)0, (C), false, false)

// ---------------------------------------------------------------------------
// Kernel 1: per-row LayerNorm over F=32, then k = xn@Wk^T, v = xn@Wv^T via WMMA.
// One wave per 16-row tile; each lane privately holds a full 32-elem row, so
// LN needs no cross-lane reduction. A-fragment selection uses a 2-way
// compile-time-indexed select on kh (no dynamic register indexing).
// ---------------------------------------------------------------------------
__global__ __launch_bounds__(32) void ln_kv_kernel(
    const float* __restrict__ x, const float* __restrict__ Wk,
    const float* __restrict__ Wv, const float* __restrict__ g,
    const float* __restrict__ bta, float* __restrict__ k_ws,
    float* __restrict__ v_ws) {
  const int lane = threadIdx.x;
  const int m = lane & 15;
  const int kh = lane >> 4;
  const int n = m;
  const long row0 = (long)blockIdx.x * 16;

  float xr[F_];
  const float* xrow = x + (row0 + m) * F_;
#pragma unroll
  for (int f = 0; f < F_; f += 4) {
    float4 t = *(const float4*)(xrow + f);
    xr[f] = t.x; xr[f + 1] = t.y; xr[f + 2] = t.z; xr[f + 3] = t.w;
  }
  float mu = 0.f;
#pragma unroll
  for (int f = 0; f < F_; ++f) mu += xr[f];
  mu *= (1.0f / F_);
  float var = 0.f;
#pragma unroll
  for (int f = 0; f < F_; ++f) { float d = xr[f] - mu; var += d * d; }
  var *= (1.0f / F_);
  const float rs = rsqrtf(var + EPS_LN_);
#pragma unroll
  for (int f = 0; f < F_; ++f) xr[f] = (xr[f] - mu) * rs * g[f] + bta[f];

  // Pre-build the 8 A fragments once (compile-time reg indices, 2-way select).
  v2f afrag[8];
#pragma unroll
  for (int c = 0; c < 8; ++c) {
    afrag[c].x = kh ? xr[4 * c + 2] : xr[4 * c];
    afrag[c].y = kh ? xr[4 * c + 3] : xr[4 * c + 1];
  }

#pragma unroll
  for (int which = 0; which < 2; ++which) {
    const float* W = which ? Wv : Wk;           // [D, F] row-major
    float* o = which ? v_ws : k_ws;             // [rows, D]
#pragma unroll
    for (int nt = 0; nt < 4; ++nt) {
      v8f acc = {};
#pragma unroll
      for (int c = 0; c < 8; ++c) {
        const int kk = 4 * c + 2 * kh;
        const float* wrow = W + (nt * 16 + n) * F_ + kk;  // B[kk][n]=W[n_glob][kk]
        v2f bf; bf.x = wrow[0]; bf.y = wrow[1];
        acc = WMMA_F32_16X16X4(afrag[c], bf, acc);
      }
#pragma unroll
      for (int r = 0; r < 8; ++r) {
        const int rowm = r + 8 * kh;
        o[(row0 + rowm) * D_ + nt * 16 + n] = acc[r];
      }
    }
  }
}

// ---------------------------------------------------------------------------
// Kernel 2: slots = mu + |sigma| * noise
// ---------------------------------------------------------------------------
__global__ void init_slots_kernel(const float* __restrict__ noise,
                                  const float* __restrict__ mu,
                                  const float* __restrict__ sigma,
                                  float* __restrict__ slots) {
  int i = blockIdx.x * blockDim.x + threadIdx.x;
  if (i < B_ * K_ * D_) {
    int d = i & (D_ - 1);
    slots[i] = mu[d] + fabsf(sigma[d]) * noise[i];
  }
}

__global__ void zero_kernel(float* __restrict__ p, int nelem) {
  int i = blockIdx.x * blockDim.x + threadIdx.x;
  if (i < nelem) p[i] = 0.f;
}

// ---------------------------------------------------------------------------
// Kernel 3: q = LN(slots) @ Wq^T, 16 rows/block, K=64 -> 16 WMMA steps
// ---------------------------------------------------------------------------
__global__ __launch_bounds__(32) void qproj_kernel(
    const float* __restrict__ slots, const float* __restrict__ Wq,
    const float* __restrict__ g, const float* __restrict__ bt,
    float* __restrict__ q_ws) {
  const int lane = threadIdx.x;
  const int m = lane & 15, kh = lane >> 4, n = m;
  const long row0 = (long)blockIdx.x * 16;

  float sr[D_];
  const float* srow = slots + (row0 + m) * D_;
#pragma unroll
  for (int d = 0; d < D_; d += 4) {
    float4 t = *(const float4*)(srow + d);
    sr[d] = t.x; sr[d + 1] = t.y; sr[d + 2] = t.z; sr[d + 3] = t.w;
  }
  float mu = 0.f;
#pragma unroll
  for (int d = 0; d < D_; ++d) mu += sr[d];
  mu *= (1.0f / D_);
  float var = 0.f;
#pragma unroll
  for (int d = 0; d < D_; ++d) { float dd = sr[d] - mu; var += dd * dd; }
  var *= (1.0f / D_);
  const float rs = rsqrtf(var + EPS_LN_);
#pragma unroll
  for (int d = 0; d < D_; ++d) sr[d] = (sr[d] - mu) * rs * g[d] + bt[d];

  // 16 A fragments, compile-time reg indices with a 2-way kh select.
  v2f afrag[16];
#pragma unroll
  for (int c = 0; c < 16; ++c) {
    afrag[c].x = kh ? sr[4 * c + 2] : sr[4 * c];
    afrag[c].y = kh ? sr[4 * c + 3] : sr[4 * c + 1];
  }

#pragma unroll
  for (int nt = 0; nt < 4; ++nt) {
    v8f acc = {};
#pragma unroll
    for (int c = 0; c < 16; ++c) {
      const int kk = 4 * c + 2 * kh;
      const float* wrow = Wq + (nt * 16 + n) * D_ + kk;
      v2f bf; bf.x = wrow[0]; bf.y = wrow[1];
      acc = WMMA_F32_16X16X4(afrag[c], bf, acc);
    }
#pragma unroll
    for (int r = 0; r < 8; ++r) {
      const int rowm = r + 8 * kh;
      q_ws[(row0 + rowm) * D_ + nt * 16 + n] = acc[r];
    }
  }
}

// ---------------------------------------------------------------------------
// Kernel 4: dots = q@k^T * SCALE (M = 8 slots padded to 16, K = 64).
// The 16 KB k-tile for this block is CONTIGUOUS in memory, so it is staged
// into LDS via the gfx1250 async-to-LDS path (ASYNCcnt), overlapped with the
// q-fragment loads. LDS rows are padded to 68 floats so B-fragment ds_load_b64
// reads hit banks (4n + 4c + 2kh) mod 64 -> conflict-free across the wave.
// Softmax over slots is lane-local (lanes 0..15 hold all 8 slot values of one
// pixel in their 8 C-VGPRs); per-(b,slot) pixel sums via wave-reduce + atomic.
// ---------------------------------------------------------------------------
#define KT_STRIDE 68
__global__ __launch_bounds__(32) void dots_kernel(
    const float* __restrict__ q_ws, const float* __restrict__ k_ws,
    float* __restrict__ attn, float* __restrict__ sums) {
  __shared__ float ktile[64 * KT_STRIDE];   // 64 pixels x (64+4) floats
  const int lane = threadIdx.x;
  const int m = lane & 15, kh = lane >> 4, n = m;
  const int b = blockIdx.x >> 6;            // 64 blocks of 64 pixels per batch
  const int p_base = (blockIdx.x & 63) * 64;

  // Async-stage the 64x64 f32 k-tile: 1024 b128 chunks, 32 per lane, writing
  // into padded LDS rows (16B chunk s of pixel-row r).
  {
    const float* gk = k_ws + ((long)b * P_ + p_base) * D_;
    const unsigned lds0 = (unsigned)(size_t)(&ktile[0]);
#pragma unroll
    for (int i = 0; i < 32; ++i) {
      const int idx = i * 32 + lane;        // chunk id 0..1023
      const int r = idx >> 4;               // pixel row 0..63
      const int s = idx & 15;               // 16B segment 0..15
      const unsigned lo = lds0 + (unsigned)(r * KT_STRIDE + s * 4) * 4u;
      const float* ga = gk + idx * 4;
      asm volatile("global_load_async_to_lds_b128 %0, %1, off"
                   :: "v"(lo), "v"(ga)
                   : "memory");
    }
  }

  // Load the q A-fragments while the async copy is in flight.
  v2f aq[16];
#pragma unroll
  for (int c = 0; c < 16; ++c) {
    const int kk = 4 * c + 2 * kh;
    if (m < K_) {
      const float* qr = q_ws + (b * K_ + m) * D_ + kk;
      aq[c].x = qr[0]; aq[c].y = qr[1];
    } else {
      aq[c].x = 0.f; aq[c].y = 0.f;         // padded slot rows 8..15
    }
  }

  float lsum[K_];
#pragma unroll
  for (int r = 0; r < K_; ++r) lsum[r] = 0.f;

  // Wait for all async LDS writes before consuming them (single wave -> no
  // barrier needed, just the ASYNCcnt wait).
  asm volatile("s_wait_asynccnt 0x0" ::: "memory");

  for (int pt = 0; pt < 4; ++pt) {
    const int p0 = p_base + pt * 16;
    v8f acc = {};
#pragma unroll
    for (int c = 0; c < 16; ++c) {
      const int kk = 4 * c + 2 * kh;
      const float* kr = &ktile[(pt * 16 + n) * KT_STRIDE + kk];
      v2f bf; bf.x = kr[0]; bf.y = kr[1];   // ds_load_b64, conflict-free
      acc = WMMA_F32_16X16X4(aq[c], bf, acc);
    }
    if (kh == 0) {  // lanes 0..15 hold slot rows 0..7 for pixel p0+n
      float mx = -1e30f;
#pragma unroll
      for (int r = 0; r < K_; ++r) mx = fmaxf(mx, acc[r] * SCALE_);
      float e[K_]; float den = 0.f;
#pragma unroll
      for (int r = 0; r < K_; ++r) { e[r] = __expf(acc[r] * SCALE_ - mx); den += e[r]; }
      const float inv = 1.0f / den;
#pragma unroll
      for (int r = 0; r < K_; ++r) {
        const float a_un = e[r] * inv + EPS_ATTN_;
        attn[((long)b * K_ + r) * P_ + p0 + n] = a_un;
        lsum[r] += a_un;
      }
    }
  }
#pragma unroll
  for (int r = 0; r < K_; ++r) {
    float v = lsum[r];                     // kh==1 lanes contribute zeros
    v += __shfl_xor(v, 1, 32);
    v += __shfl_xor(v, 2, 32);
    v += __shfl_xor(v, 4, 32);
    v += __shfl_xor(v, 8, 32);
    v += __shfl_xor(v, 16, 32);
    if (lane == 0) atomicAdd(&sums[b * K_ + r], v);
  }
}

// ---------------------------------------------------------------------------
// Kernel 5: updates[b,s,:] = (sum_p attn_un[b,s,p] * v[b,p,:]) / sums[b,s].
// One block per batch, 8 waves each accumulate a 512-pixel K-range via WMMA,
// partials merged through LDS float atomics.
// ---------------------------------------------------------------------------
__global__ __launch_bounds__(256) void updates_kernel(
    const float* __restrict__ attn, const float* __restrict__ v_ws,
    const float* __restrict__ sums, float* __restrict__ upd) {
  const int b = blockIdx.x;
  const int lane = threadIdx.x & 31;
  const int w = threadIdx.x >> 5;           // wave 0..7
  const int m = lane & 15, kh = lane >> 4, n = m;

  __shared__ float red[4 * 16 * 16];
  for (int i = threadIdx.x; i < 1024; i += 256) red[i] = 0.f;
  __syncthreads();

  v8f acc0 = {}, acc1 = {}, acc2 = {}, acc3 = {};
  const long vb = (long)b * P_;
  for (int ch = 0; ch < 128; ++ch) {
    const int pc = w * 512 + ch * 4;
    const int kk = pc + 2 * kh;
    v2f a;
    if (m < K_) {
      const float* ar = attn + ((long)b * K_ + m) * P_ + kk;
      a.x = ar[0]; a.y = ar[1];
    } else {
      a.x = 0.f; a.y = 0.f;
    }
    const float* vr0 = v_ws + (vb + kk) * D_;  // B[kk][n] = v[kk][col]
    const float* vr1 = vr0 + D_;
    v2f bf;
    bf.x = vr0[0 * 16 + n]; bf.y = vr1[0 * 16 + n]; acc0 = WMMA_F32_16X16X4(a, bf, acc0);
    bf.x = vr0[1 * 16 + n]; bf.y = vr1[1 * 16 + n]; acc1 = WMMA_F32_16X16X4(a, bf, acc1);
    bf.x = vr0[2 * 16 + n]; bf.y = vr1[2 * 16 + n]; acc2 = WMMA_F32_16X16X4(a, bf, acc2);
    bf.x = vr0[3 * 16 + n]; bf.y = vr1[3 * 16 + n]; acc3 = WMMA_F32_16X16X4(a, bf, acc3);
  }
#pragma unroll
  for (int r = 0; r < 8; ++r) {
    const int row = r + 8 * kh;
    atomicAdd(&red[0 * 256 + row * 16 + n], acc0[r]);
    atomicAdd(&red[1 * 256 + row * 16 + n], acc1[r]);
    atomicAdd(&red[2 * 256 + row * 16 + n], acc2[r]);
    atomicAdd(&red[3 * 256 + row * 16 + n], acc3[r]);
  }
  __syncthreads();
  for (int i = threadIdx.x; i < 1024; i += 256) {
    const int nt = i >> 8, rem = i & 255, row = rem >> 4, col = rem & 15;
    if (row < K_) {
      const float s = sums[b * K_ + row];
      upd[((long)b * K_ + row) * D_ + nt * 16 + col] = red[i] / s;
    }
  }
}

// ---------------------------------------------------------------------------
// Kernel 6: GRUCell + LN + FFN residual for one (b,slot) row per block.
// Tiny (1024 rows x ~1K MACs) -> plain VALU.
// ---------------------------------------------------------------------------
__global__ __launch_bounds__(64) void gru_ff_kernel(
    const float* __restrict__ upd, const float* __restrict__ slots_in,
    const float* __restrict__ W_ih, const float* __restrict__ W_hh,
    const float* __restrict__ b_ih, const float* __restrict__ b_hh,
    const float* __restrict__ ln_g, const float* __restrict__ ln_b,
    const float* __restrict__ W1, const float* __restrict__ b1,
    const float* __restrict__ W2, const float* __restrict__ b2,
    float* __restrict__ slots_out) {
  const int row = blockIdx.x;
  const int t = threadIdx.x;
  __shared__ float xu[D_], hp[D_], hn_s[D_], ffin[D_], ff1[2 * D_];
  xu[t] = upd[row * D_ + t];
  hp[t] = slots_in[row * D_ + t];
  __syncthreads();
  float gi_r = b_ih[t], gi_z = b_ih[D_ + t], gi_n = b_ih[2 * D_ + t];
  float gh_r = b_hh[t], gh_z = b_hh[D_ + t], gh_n = b_hh[2 * D_ + t];
  for (int j = 0; j < D_; ++j) {
    const float xj = xu[j], hj = hp[j];
    gi_r += W_ih[t * D_ + j] * xj;
    gi_z += W_ih[(D_ + t) * D_ + j] * xj;
    gi_n += W_ih[(2 * D_ + t) * D_ + j] * xj;
    gh_r += W_hh[t * D_ + j] * hj;
    gh_z += W_hh[(D_ + t) * D_ + j] * hj;
    gh_n += W_hh[(2 * D_ + t) * D_ + j] * hj;
  }
  const float r = 1.f / (1.f + __expf(-(gi_r + gh_r)));
  const float z = 1.f / (1.f + __expf(-(gi_z + gh_z)));
  const float nn = tanhf(gi_n + r * gh_n);
  const float hn = (1.f - z) * nn + z * hp[t];
  hn_s[t] = hn;
  __syncthreads();
  float mu = 0.f;
  for (int j = 0; j < D_; ++j) mu += hn_s[j];
  mu *= (1.f / D_);
  float var = 0.f;
  for (int j = 0; j < D_; ++j) { const float d = hn_s[j] - mu; var += d * d; }
  var *= (1.f / D_);
  ffin[t] = (hn - mu) * rsqrtf(var + EPS_LN_) * ln_g[t] + ln_b[t];
  __syncthreads();
  float a0 = b1[t], a1 = b1[D_ + t];
  for (int j = 0; j < D_; ++j) {
    a0 += W1[t * D_ + j] * ffin[j];
    a1 += W1[(D_ + t) * D_ + j] * ffin[j];
  }
  ff1[t] = fmaxf(a0, 0.f);
  ff1[D_ + t] = fmaxf(a1, 0.f);
  __syncthreads();
  float o = b2[t];
  for (int j = 0; j < 2 * D_; ++j) o += W2[t * 2 * D_ + j] * ff1[j];
  slots_out[row * D_ + t] = hn + o;
}

// ---------------------------------------------------------------------------
// Kernel 7: final attn normalization into d_out
// ---------------------------------------------------------------------------
__global__ void attn_norm_kernel(const float* __restrict__ attn,
                                 const float* __restrict__ sums,
                                 float* __restrict__ out) {
  const long i = (long)blockIdx.x * blockDim.x + threadIdx.x;
  if (i < (long)B_ * K_ * P_) {
    const long bk = i / P_;
    out[i] = attn[i] / sums[bk];
  }
}

extern "C" void kernel_launch(void* const* d_in, const int* in_sizes, int n_in,
                              void* d_out, int out_size, void* d_ws, size_t ws_size,
                              hipStream_t stream) {
  (void)in_sizes; (void)n_in; (void)out_size; (void)ws_size;
  const float* x       = (const float*)d_in[0];
  const float* noise   = (const float*)d_in[1];
  const float* mu      = (const float*)d_in[2];
  const float* sigma   = (const float*)d_in[3];
  const float* Wq      = (const float*)d_in[4];
  const float* Wk      = (const float*)d_in[5];
  const float* Wv      = (const float*)d_in[6];
  const float* W_ih    = (const float*)d_in[7];
  const float* W_hh    = (const float*)d_in[8];
  const float* b_ih    = (const float*)d_in[9];
  const float* b_hh    = (const float*)d_in[10];
  const float* ln_in_g = (const float*)d_in[11];
  const float* ln_in_b = (const float*)d_in[12];
  const float* ln_s_g  = (const float*)d_in[13];
  const float* ln_s_b  = (const float*)d_in[14];
  const float* ff_ln_g = (const float*)d_in[15];
  const float* ff_ln_b = (const float*)d_in[16];
  const float* ff_W1   = (const float*)d_in[17];
  const float* ff_b1   = (const float*)d_in[18];
  const float* ff_W2   = (const float*)d_in[19];
  const float* ff_b2   = (const float*)d_in[20];

  float* ws = (float*)d_ws;
  size_t off = 0;
  float* k_ws    = ws + off; off += (size_t)B_ * P_ * D_;
  float* v_ws    = ws + off; off += (size_t)B_ * P_ * D_;
  float* attn_ws = ws + off; off += (size_t)B_ * K_ * P_;
  float* slots   = ws + off; off += (size_t)B_ * K_ * D_;
  float* q_ws    = ws + off; off += (size_t)B_ * K_ * D_;
  float* upd     = ws + off; off += (size_t)B_ * K_ * D_;
  float* sums    = ws + off; off += (size_t)B_ * K_;

  float* out_slots = (float*)d_out;
  float* out_attn  = out_slots + (size_t)B_ * K_ * D_;

  ln_kv_kernel<<<(B_ * P_) / 16, 32, 0, stream>>>(x, Wk, Wv, ln_in_g, ln_in_b,
                                                  k_ws, v_ws);
  init_slots_kernel<<<(B_ * K_ * D_ + 255) / 256, 256, 0, stream>>>(noise, mu,
                                                                    sigma, slots);
  for (int it = 0; it < 2; ++it) {
    qproj_kernel<<<(B_ * K_) / 16, 32, 0, stream>>>(slots, Wq, ln_s_g, ln_s_b, q_ws);
    zero_kernel<<<(B_ * K_ + 255) / 256, 256, 0, stream>>>(sums, B_ * K_);
    dots_kernel<<<B_ * (P_ / 64), 32, 0, stream>>>(q_ws, k_ws, attn_ws, sums);
    updates_kernel<<<B_, 256, 0, stream>>>(attn_ws, v_ws, sums, upd);
    gru_ff_kernel<<<B_ * K_, 64, 0, stream>>>(upd, slots, W_ih, W_hh, b_ih, b_hh,
                                              ff_ln_g, ff_ln_b, ff_W1, ff_b1,
                                              ff_W2, ff_b2,
                                              (it == 1) ? out_slots : slots);
  }
  attn_norm_kernel<<<((long)B_ * K_ * P_ + 255) / 256, 256, 0, stream>>>(
      attn_ws, sums, out_attn);
}